// GATv4Conv_74131135529464
// MI455X (gfx1250) — compile-verified
//
#include <hip/hip_runtime.h>
#include <hip/hip_bf16.h>
#include <math.h>

// ---------------------------------------------------------------------------
// GATv4Conv for MI455X (gfx1250, wave32).
//   Phase P: node projections via V_WMMA_F32_16X16X4_F32 (fp32 matrix core)
//   Phase E: edge gather/score/softmax/aggregate, bandwidth + L2-atomic bound
// ---------------------------------------------------------------------------

#define IN_DIM   128
#define HF       128   // H*F
#define NHEAD    4
#define FDIM     32
#define OUT_LDC  160   // (H+1)*F
#define NEG_SLOPE 0.2f

typedef __attribute__((ext_vector_type(2))) float v2f;
typedef __attribute__((ext_vector_type(8))) float v8f;

// ---- helpers --------------------------------------------------------------

__device__ __forceinline__ float atomAddF(float* p, float v) {
    return __hip_atomic_fetch_add(p, v, __ATOMIC_RELAXED, __HIP_MEMORY_SCOPE_AGENT);
}
__device__ __forceinline__ unsigned atomMaxU(unsigned* p, unsigned v) {
    return __hip_atomic_fetch_max(p, v, __ATOMIC_RELAXED, __HIP_MEMORY_SCOPE_AGENT);
}
// order-preserving float<->uint encoding so atomic-umax == float max
__device__ __forceinline__ unsigned encF(float f) {
    unsigned u = __float_as_uint(f);
    return (u & 0x80000000u) ? ~u : (u | 0x80000000u);
}
__device__ __forceinline__ float decF(unsigned u) {
    return (u & 0x80000000u) ? __uint_as_float(u ^ 0x80000000u)
                             : __uint_as_float(~u);
}
#define SMAX_INIT 0x007FFFFFu   // encF(-inf)

// ---- init: zero ft region of d_out, init smax/denom -----------------------

__global__ void init_kernel(float* __restrict__ out, unsigned* __restrict__ smax,
                            float* __restrict__ denom, int n) {
    long long idx = (long long)blockIdx.x * blockDim.x + threadIdx.x;
    long long tot = (long long)n * HF;
    if (idx < tot) {
        long long node = idx >> 7;          // /128
        int off = (int)(idx & 127);
        out[node * OUT_LDC + FDIM + off] = 0.0f;
    }
    if (idx < (long long)n * NHEAD) {
        smax[idx]  = SMAX_INIT;
        denom[idx] = 0.0f;
    }
}

// ---- projection GEMM: out[M,Ncol] = X[M,128] @ W[128,Ncol] + bias ---------
// one wave per 16x16 output tile; 32 x V_WMMA_F32_16X16X4_F32 along K=128.

__global__ void gemm_wmma_f32(const float* __restrict__ X,
                              const float* __restrict__ W,
                              const float* __restrict__ bias,
                              float* __restrict__ out,
                              int K, int Ncol, int ldc) {
    const int lane = threadIdx.x;          // 0..31, blockDim.x == 32
    const int colT = blockIdx.x;           // Ncol/16 tiles
    const int rowT = blockIdx.y;           // M/16 tiles
    const int m    = lane & 15;
    const int kq   = lane >> 4;            // 0: K={0,1}, 1: K={2,3}
    const int col  = colT * 16 + m;

    const float* __restrict__ Xrow = X + (size_t)(rowT * 16 + m) * K;

    v8f acc = {};
    #pragma unroll 4
    for (int k0 = 0; k0 < K; k0 += 4) {
        v2f a, b;
        a.x = Xrow[k0 + 2 * kq];
        a.y = Xrow[k0 + 2 * kq + 1];
        b.x = W[(size_t)(k0 + 2 * kq)     * Ncol + col];
        b.y = W[(size_t)(k0 + 2 * kq + 1) * Ncol + col];
        acc = __builtin_amdgcn_wmma_f32_16x16x4_f32(
                  /*neg_a=*/false, a, /*neg_b=*/false, b,
                  /*c_mod=*/(short)0, acc,
                  /*reuse_a=*/false, /*reuse_b=*/false);
    }

    const float bv = bias[col];
    #pragma unroll
    for (int v = 0; v < 8; ++v) {
        int row = rowT * 16 + v + 8 * kq;   // C/D layout: lanes16-31 hold M+8
        out[(size_t)row * ldc + col] = acc[v] + bv;
    }
}

// ---- edge scores + segment max --------------------------------------------
// one wave per edge: lane L loads float4 at offset 4L of the 128-float rows.

__global__ void edge_score_kernel(const float* __restrict__ el_mut,
                                  const float* __restrict__ er_mut,
                                  const float* __restrict__ attn,
                                  const int* __restrict__ src,
                                  const int* __restrict__ dst,
                                  float* __restrict__ s_buf,
                                  unsigned* __restrict__ smax,
                                  int E) {
    const int lane = threadIdx.x & 31;
    const long long e = (long long)blockIdx.x * (blockDim.x >> 5) + (threadIdx.x >> 5);
    if (e >= E) return;

    const int si = src[e];
    const int di = dst[e];

    const float4 a = ((const float4*)(el_mut + (size_t)si * HF))[lane];
    const float4 b = ((const float4*)(er_mut + (size_t)di * HF))[lane];
    const float4 w = ((const float4*)attn)[lane];

    float4 t;
    t.x = a.x + b.x; t.y = a.y + b.y; t.z = a.z + b.z; t.w = a.w + b.w;
    t.x = t.x > 0.f ? t.x : NEG_SLOPE * t.x;
    t.y = t.y > 0.f ? t.y : NEG_SLOPE * t.y;
    t.z = t.z > 0.f ? t.z : NEG_SLOPE * t.z;
    t.w = t.w > 0.f ? t.w : NEG_SLOPE * t.w;

    float p = t.x * w.x + t.y * w.y + t.z * w.z + t.w * w.w;
    // reduce within each 8-lane head group (lanes 8h..8h+7 cover head h)
    p += __shfl_down(p, 4, 8);
    p += __shfl_down(p, 2, 8);
    p += __shfl_down(p, 1, 8);

    if ((lane & 7) == 0) {
        const int h = lane >> 3;
        s_buf[e * NHEAD + h] = p;
        atomMaxU(&smax[(size_t)di * NHEAD + h], encF(p));
    }
}

// ---- exp(s - smax[dst]) and segment-sum denominator -----------------------

__global__ void edge_exp_kernel(const int* __restrict__ dst,
                                const unsigned* __restrict__ smax,
                                float* __restrict__ s_buf,   // in: s, out: ex
                                float* __restrict__ denom,
                                long long total) {
    long long idx = (long long)blockIdx.x * blockDim.x + threadIdx.x;
    if (idx >= total) return;
    long long e = idx >> 2;
    int h = (int)(idx & 3);
    int di = dst[e];
    float mx = decF(smax[(size_t)di * NHEAD + h]);
    float ex = __expf(s_buf[idx] - mx);
    s_buf[idx] = ex;
    atomAddF(&denom[(size_t)di * NHEAD + h], ex);
}

// ---- normalize + weighted aggregation into d_out (ft region) --------------

__global__ void edge_agg_kernel(const float* __restrict__ el_self,
                                const int* __restrict__ src,
                                const int* __restrict__ dst,
                                const float* __restrict__ ex_buf,
                                const float* __restrict__ denom,
                                float* __restrict__ out,
                                int E) {
    const int lane = threadIdx.x & 31;
    const long long e = (long long)blockIdx.x * (blockDim.x >> 5) + (threadIdx.x >> 5);
    if (e >= E) return;

    const int si = src[e];
    const int di = dst[e];
    const int h  = lane >> 3;                 // 4 floats/lane => head = lane/8

    const float alpha = ex_buf[e * NHEAD + h] /
                        denom[(size_t)di * NHEAD + h];

    const float4 v = ((const float4*)(el_self + (size_t)si * HF))[lane];
    float* op = out + (size_t)di * OUT_LDC + FDIM + lane * 4;
    atomAddF(op + 0, v.x * alpha);
    atomAddF(op + 1, v.y * alpha);
    atomAddF(op + 2, v.z * alpha);
    atomAddF(op + 3, v.w * alpha);
}

// ---------------------------------------------------------------------------

extern "C" void kernel_launch(void* const* d_in, const int* in_sizes, int n_in,
                              void* d_out, int out_size, void* d_ws, size_t ws_size,
                              hipStream_t stream) {
    const float* feat   = (const float*)d_in[0];
    const float* W_src  = (const float*)d_in[1];
    const float* b_src  = (const float*)d_in[2];
    const float* W_dst  = (const float*)d_in[3];
    const float* b_dst  = (const float*)d_in[4];
    const float* W_self = (const float*)d_in[5];
    const float* b_self = (const float*)d_in[6];
    const float* W_lin  = (const float*)d_in[7];
    const float* b_lin  = (const float*)d_in[8];
    const float* attn   = (const float*)d_in[9];
    const int*   src    = (const int*)d_in[10];
    const int*   dst    = (const int*)d_in[11];

    const int N = in_sizes[0] / IN_DIM;     // 50000
    const int E = in_sizes[10];             // 800000

    float* out = (float*)d_out;

    // workspace layout (floats)
    const size_t nPr = (size_t)N * HF;
    float*    el_mut  = (float*)d_ws;
    float*    er_mut  = el_mut + nPr;
    float*    el_self = er_mut + nPr;
    float*    s_buf   = el_self + nPr;                    // E*NHEAD
    unsigned* smax    = (unsigned*)(s_buf + (size_t)E * NHEAD);  // N*NHEAD
    float*    denom   = (float*)(smax + (size_t)N * NHEAD);      // N*NHEAD

    // 1) init output ft region + softmax state
    {
        long long tot = (long long)N * HF;
        int grid = (int)((tot + 255) / 256);
        init_kernel<<<grid, 256, 0, stream>>>(out, smax, denom, N);
    }

    // 2) node projections (WMMA fp32). M=50000 -> 3125 row tiles exactly.
    {
        dim3 blk(32, 1, 1);
        dim3 g128(HF / 16, N / 16, 1);
        gemm_wmma_f32<<<g128, blk, 0, stream>>>(feat, W_src,  b_src,  el_mut,  IN_DIM, HF, HF);
        gemm_wmma_f32<<<g128, blk, 0, stream>>>(feat, W_dst,  b_dst,  er_mut,  IN_DIM, HF, HF);
        gemm_wmma_f32<<<g128, blk, 0, stream>>>(feat, W_self, b_self, el_self, IN_DIM, HF, HF);
        dim3 g32(FDIM / 16, N / 16, 1);
        // feat_lin written straight into d_out[:, 0, :] with ldc = 160
        gemm_wmma_f32<<<g32, blk, 0, stream>>>(feat, W_lin, b_lin, out, IN_DIM, FDIM, OUT_LDC);
    }

    // 3) edge scores + segment max (wave per edge, 8 edges / 256-thr block)
    {
        int grid = (E + 7) / 8;
        edge_score_kernel<<<grid, 256, 0, stream>>>(el_mut, er_mut, attn, src, dst,
                                                    s_buf, smax, E);
    }

    // 4) exp + denominator
    {
        long long tot = (long long)E * NHEAD;
        int grid = (int)((tot + 255) / 256);
        edge_exp_kernel<<<grid, 256, 0, stream>>>(dst, smax, s_buf, denom, tot);
    }

    // 5) normalize + aggregate into d_out ft region
    {
        int grid = (E + 7) / 8;
        edge_agg_kernel<<<grid, 256, 0, stream>>>(el_self, src, dst, s_buf, denom,
                                                  out, E);
    }
}